// LSTM_VAE_8950711845787
// MI455X (gfx1250) — compile-verified
//
#include <hip/hip_runtime.h>
#include <hip/hip_bf16.h>
#include <math.h>

typedef float v2f __attribute__((ext_vector_type(2)));
typedef float v8f __attribute__((ext_vector_type(8)));

#define T_STEPS 131072
#define FEATS 128
#define HID 64
#define LAT 16
#define NGATES 256   // 4*HID

// ---------------------------------------------------------------------------
// Kernel 1: G[t, j] = sum_k X[t,k] * W_ih[j,k] + (b_ih[j] + b_hh[j])
// f32 WMMA GEMM. Each wave owns a 16x256 row stripe of G: the 32 A fragments
// (16x128 stripe of X) are loaded ONCE into VGPRs, then reused across all 16
// column tiles (B = W_ih stays hot in WGP$/L2). HBM traffic = compulsory
// 64MB read + 128MB write ~= bandwidth floor (~8us @ 23.3 TB/s).
// A (16x4 f32) layout: lanes 0-15 = M rows, VGPR0/1 = K {0,1}; lanes 16-31 K {2,3}.
// ---------------------------------------------------------------------------
__global__ __launch_bounds__(256) void gates_gemm_wmma(
    const float* __restrict__ X, const float* __restrict__ Wih,
    const float* __restrict__ bih, const float* __restrict__ bhh,
    float* __restrict__ G) {
  const int lane  = threadIdx.x & 31;
  const int wave  = threadIdx.x >> 5;
  const int tileM = blockIdx.x * 8 + wave;   // 8192 row stripes
  const int half  = lane >> 4;
  const int lq    = lane & 15;

  // Load the whole 16x128 A stripe into registers (64 VGPRs/lane).
  v2f afrag[32];
  {
    const float* ap = X + (tileM * 16 + lq) * FEATS + 2 * half;
#pragma unroll
    for (int k0 = 0; k0 < FEATS; k0 += 4)
      afrag[k0 >> 2] = *(const v2f*)(ap + k0);   // A[M=lq][k0 + 2*half + {0,1}]
  }

  float* grow = G + (tileM * 16 + 8 * half) * NGATES;

  for (int tn = 0; tn < 16; ++tn) {            // 16 column tiles of 16
    const float* bp = Wih + (tn * 16 + lq) * FEATS + 2 * half;
    v8f acc = {};
#pragma unroll
    for (int k0 = 0; k0 < FEATS; k0 += 4) {
      v2f b = *(const v2f*)(bp + k0);          // B[k][N=lq] = Wih[N][k]
      acc = __builtin_amdgcn_wmma_f32_16x16x4_f32(false, afrag[k0 >> 2], false, b,
                                                  (short)0, acc, false, false);
    }
    const int   col = tn * 16 + lq;
    const float bj  = bih[col] + bhh[col];
    float* gout = grow + col;
#pragma unroll
    for (int r = 0; r < 8; ++r) gout[r * NGATES] = acc[r] + bj;  // row M = r + 8*half
  }
}

// ---------------------------------------------------------------------------
// Kernel 2: sequential LSTM-VAE scan. One persistent 256-thread workgroup.
// W_hh rows live in VGPRs (thread j owns gate row j); all other weights in
// LDS with padded row strides (bank-conflict-free column reads).
// ---------------------------------------------------------------------------
// LDS float offsets
#define OWe1 0
#define OWe2 (OWe1 + 64 * 65)
#define OWd2 (OWe2 + 64 * 65)
#define OWe3 (OWd2 + 64 * 65)           // 32x65
#define OWd1 (OWe3 + 32 * 65)           // 64x17
#define OWd3 (OWd1 + 64 * 17)           // 128x65
#define Obe1 (OWd3 + 128 * 65)
#define Obe2 (Obe1 + 64)
#define Obe3 (Obe2 + 64)                // 32
#define Obd1 (Obe3 + 32)
#define Obd2 (Obd1 + 64)
#define Obd3 (Obd2 + 64)                // 128
#define Oh   (Obd3 + 128)               // 64 (16B aligned: offset %4 == 0)
#define Ogb  (Oh + 64)                  // 256 gates
#define Oe1  (Ogb + 256)
#define Oe2  (Oe1 + 64)
#define Oml  (Oe2 + 64)                 // 32
#define Oz   (Oml + 32)                 // 16
#define Od1  (Oz + 16)
#define Od2  (Od1 + 64)
#define SMEM_FLOATS (Od2 + 64)          // 25008 floats = 100032 bytes

__device__ __forceinline__ float sigm(float x) { return 1.0f / (1.0f + __expf(-x)); }
__device__ __forceinline__ float prelu(float x, float a) { return x > 0.0f ? x : a * x; }

__global__ __launch_bounds__(256) void lstm_vae_scan(
    const float* __restrict__ G,   const float* __restrict__ eps,
    const float* __restrict__ h0,  const float* __restrict__ c0,
    const int*   __restrict__ trn, const float* __restrict__ Whh,
    const float* __restrict__ We1, const float* __restrict__ be1, const float* __restrict__ ae1,
    const float* __restrict__ We2, const float* __restrict__ be2, const float* __restrict__ ae2,
    const float* __restrict__ We3, const float* __restrict__ be3,
    const float* __restrict__ Wd1, const float* __restrict__ bd1, const float* __restrict__ ad1,
    const float* __restrict__ Wd2, const float* __restrict__ bd2, const float* __restrict__ ad2,
    const float* __restrict__ Wd3, const float* __restrict__ bd3,
    float* __restrict__ outs, float* __restrict__ mus, float* __restrict__ lvs) {
  extern __shared__ float sm[];
  const int tid = threadIdx.x;

  // ---- stage weights into LDS (padded rows) ----
  for (int i = tid; i < 64 * 64; i += 256) sm[OWe1 + (i >> 6) * 65 + (i & 63)] = We1[i];
  for (int i = tid; i < 64 * 64; i += 256) sm[OWe2 + (i >> 6) * 65 + (i & 63)] = We2[i];
  for (int i = tid; i < 64 * 64; i += 256) sm[OWd2 + (i >> 6) * 65 + (i & 63)] = Wd2[i];
  for (int i = tid; i < 32 * 64; i += 256) sm[OWe3 + (i >> 6) * 65 + (i & 63)] = We3[i];
  for (int i = tid; i < 64 * 16; i += 256) sm[OWd1 + (i >> 4) * 17 + (i & 15)] = Wd1[i];
  for (int i = tid; i < 128 * 64; i += 256) sm[OWd3 + (i >> 6) * 65 + (i & 63)] = Wd3[i];
  if (tid < 64)  { sm[Obe1 + tid] = be1[tid]; sm[Obe2 + tid] = be2[tid];
                   sm[Obd1 + tid] = bd1[tid]; sm[Obd2 + tid] = bd2[tid]; }
  if (tid < 32)  sm[Obe3 + tid] = be3[tid];
  if (tid < 128) sm[Obd3 + tid] = bd3[tid];

  // ---- W_hh row for this thread's gate in VGPRs ----
  float whh[64];
  {
    const float4* wp = (const float4*)(Whh + tid * 64);
#pragma unroll
    for (int q = 0; q < 16; ++q) {
      float4 v = wp[q];
      whh[4 * q + 0] = v.x; whh[4 * q + 1] = v.y;
      whh[4 * q + 2] = v.z; whh[4 * q + 3] = v.w;
    }
  }

  float creg = 0.0f;
  if (tid < HID) { sm[Oh + tid] = h0[tid]; creg = c0[tid]; }
  const int   training = *trn;
  const float vae1 = *ae1, vae2 = *ae2, vad1 = *ad1, vad2 = *ad2;
  __syncthreads();

  for (int t = 0; t < T_STEPS; ++t) {
    if (t + 16 < T_STEPS)
      __builtin_prefetch(G + (t + 16) * NGATES + tid, 0, 1);   // global_prefetch_b8

    // gates[j] = G[t,j] + h . Whh[j,:]
    float hreg[64];
    {
      const float4* hp = (const float4*)&sm[Oh];
#pragma unroll
      for (int q = 0; q < 16; ++q) {
        float4 v = hp[q];
        hreg[4 * q + 0] = v.x; hreg[4 * q + 1] = v.y;
        hreg[4 * q + 2] = v.z; hreg[4 * q + 3] = v.w;
      }
    }
    float acc = G[t * NGATES + tid];
#pragma unroll
    for (int k = 0; k < 64; ++k) acc = fmaf(hreg[k], whh[k], acc);
    __syncthreads();                 // prior readers of sm[Oh] done
    sm[Ogb + tid] = acc;
    __syncthreads();

    // LSTM cell: threads 0..63
    if (tid < HID) {
      float iv = sigm(sm[Ogb + tid]);
      float fv = sigm(sm[Ogb + 64 + tid]);
      float gv = tanhf(sm[Ogb + 128 + tid]);
      float ov = sigm(sm[Ogb + 192 + tid]);
      creg = fv * creg + iv * gv;
      sm[Oh + tid] = ov * tanhf(creg);
    }
    __syncthreads();

    // encoder layer 1
    if (tid < HID) {
      float a = sm[Obe1 + tid];
#pragma unroll
      for (int k = 0; k < 64; ++k) a = fmaf(sm[Oh + k], sm[OWe1 + tid * 65 + k], a);
      sm[Oe1 + tid] = prelu(a, vae1);
    }
    __syncthreads();
    // encoder layer 2
    if (tid < HID) {
      float a = sm[Obe2 + tid];
#pragma unroll
      for (int k = 0; k < 64; ++k) a = fmaf(sm[Oe1 + k], sm[OWe2 + tid * 65 + k], a);
      sm[Oe2 + tid] = prelu(a, vae2);
    }
    __syncthreads();
    // mu / logvar
    if (tid < 2 * LAT) {
      float a = sm[Obe3 + tid];
#pragma unroll
      for (int k = 0; k < 64; ++k) a = fmaf(sm[Oe2 + k], sm[OWe3 + tid * 65 + k], a);
      sm[Oml + tid] = a;
      if (tid < LAT) mus[t * LAT + tid] = a;
      else           lvs[t * LAT + (tid - LAT)] = a;
    }
    __syncthreads();
    // reparameterize
    if (tid < LAT) {
      float mu = sm[Oml + tid], lv = sm[Oml + LAT + tid];
      sm[Oz + tid] = training ? fmaf(eps[t * LAT + tid], __expf(0.5f * lv), mu) : mu;
    }
    __syncthreads();
    // decoder layer 1 (16 -> 64)
    if (tid < HID) {
      float a = sm[Obd1 + tid];
#pragma unroll
      for (int k = 0; k < LAT; ++k) a = fmaf(sm[Oz + k], sm[OWd1 + tid * 17 + k], a);
      sm[Od1 + tid] = prelu(a, vad1);
    }
    __syncthreads();
    // decoder layer 2 (64 -> 64)
    if (tid < HID) {
      float a = sm[Obd2 + tid];
#pragma unroll
      for (int k = 0; k < 64; ++k) a = fmaf(sm[Od1 + k], sm[OWd2 + tid * 65 + k], a);
      sm[Od2 + tid] = prelu(a, vad2);
    }
    __syncthreads();
    // decoder layer 3 (64 -> 128) + output
    if (tid < FEATS) {
      float a = sm[Obd3 + tid];
#pragma unroll
      for (int k = 0; k < 64; ++k) a = fmaf(sm[Od2 + k], sm[OWd3 + tid * 65 + k], a);
      outs[t * FEATS + tid] = 4.0f * tanhf(a);
    }
    __syncthreads();
  }
}

extern "C" void kernel_launch(void* const* d_in, const int* in_sizes, int n_in,
                              void* d_out, int out_size, void* d_ws, size_t ws_size,
                              hipStream_t stream) {
  (void)in_sizes; (void)n_in; (void)out_size; (void)ws_size;
  const float* x    = (const float*)d_in[0];
  const float* eps  = (const float*)d_in[1];
  const float* h0   = (const float*)d_in[2];
  const float* c0   = (const float*)d_in[3];
  const int*   trn  = (const int*)  d_in[4];
  const float* Wih  = (const float*)d_in[5];
  const float* Whh  = (const float*)d_in[6];
  const float* bih  = (const float*)d_in[7];
  const float* bhh  = (const float*)d_in[8];
  const float* We1  = (const float*)d_in[9];
  const float* be1  = (const float*)d_in[10];
  const float* ae1  = (const float*)d_in[11];
  const float* We2  = (const float*)d_in[12];
  const float* be2  = (const float*)d_in[13];
  const float* ae2  = (const float*)d_in[14];
  const float* We3  = (const float*)d_in[15];
  const float* be3  = (const float*)d_in[16];
  const float* Wd1  = (const float*)d_in[17];
  const float* bd1  = (const float*)d_in[18];
  const float* ad1  = (const float*)d_in[19];
  const float* Wd2  = (const float*)d_in[20];
  const float* bd2  = (const float*)d_in[21];
  const float* ad2  = (const float*)d_in[22];
  const float* Wd3  = (const float*)d_in[23];
  const float* bd3  = (const float*)d_in[24];

  float* G    = (float*)d_ws;                      // T x 256 precomputed gates (128MB)
  float* outs = (float*)d_out;                     // T x 128
  float* musp = outs + (size_t)T_STEPS * FEATS;    // T x 16
  float* lvsp = musp + (size_t)T_STEPS * LAT;      // T x 16

  // 8192 row stripes, 8 waves (stripes) per 256-thread block
  gates_gemm_wmma<<<dim3(1024), dim3(256), 0, stream>>>(x, Wih, bih, bhh, G);

  lstm_vae_scan<<<dim3(1), dim3(256), SMEM_FLOATS * sizeof(float), stream>>>(
      G, eps, h0, c0, trn, Whh,
      We1, be1, ae1, We2, be2, ae2, We3, be3,
      Wd1, bd1, ad1, Wd2, bd2, ad2, Wd3, bd3,
      outs, musp, lvsp);
}